// DistributedPagedAttention_82068235091992
// MI455X (gfx1250) — compile-verified
//
#include <hip/hip_runtime.h>
#include <stdint.h>

// ---- problem constants (match reference) ----
#define NUM_RANKS     4
#define NUM_SEQS      32
#define NUM_HEADS     32
#define NUM_KV_HEADS  8
#define GQA           4          // NUM_HEADS / NUM_KV_HEADS
#define HEAD_DIM      128
#define BLK           16         // paged block size (== WMMA tile M/N)
#define MAX_BLOCKS    64         // blocks per seq
#define NBLOCKS       2048       // blocks per rank
#define SCALE         0.08838834764831845f   // 1/sqrt(128)

#define WAVES         4          // waves per workgroup (128 threads)
#define VSTRIDE       28         // LDS row stride (words) for V tile [d][t] (B128-aligned, read-conflict-free)
#define PSTRIDE       17         // LDS row stride for P tile

typedef __attribute__((ext_vector_type(2))) float v2f;
typedef __attribute__((ext_vector_type(8))) float v8f;

__device__ __forceinline__ float wave_max(float v) {
#pragma unroll
  for (int m = 16; m >= 1; m >>= 1) v = fmaxf(v, __shfl_xor(v, m, 32));
  return v;
}
__device__ __forceinline__ float wave_sum(float v) {
#pragma unroll
  for (int m = 16; m >= 1; m >>= 1) v += __shfl_xor(v, m, 32);
  return v;
}

// CDNA5 async global->LDS copy (ASYNCcnt-tracked), 16B per lane.
// vdst VGPR holds the wave-relative LDS byte offset; vaddr is the 64-bit global address.
__device__ __forceinline__ void async_copy_b128(const float* gptr, uint32_t lds_byte_off) {
  asm volatile("global_load_async_to_lds_b128 %0, %1, off"
               :: "v"(lds_byte_off), "v"(gptr) : "memory");
}
__device__ __forceinline__ void wait_asynccnt0() {
  asm volatile("s_wait_asynccnt 0x0" ::: "memory");
}
__device__ __forceinline__ void wait_dscnt0() {
  asm volatile("s_wait_dscnt 0x0" ::: "memory");
}

// Phase 1: per (rank, seq, kv-head): flash-style attention over paged KV.
// Emits unnormalized o (sum p*V), row max m, exp-sum se into workspace.
__global__ __launch_bounds__(WAVES * 32)
void pa_rank_kernel(const float* __restrict__ query,
                    const float* __restrict__ key_cache,
                    const float* __restrict__ value_cache,
                    const int*   __restrict__ block_tables,
                    const int*   __restrict__ context_lens,
                    float* __restrict__ o_un,
                    float* __restrict__ m_out,
                    float* __restrict__ se_out)
{
  const int kh   = blockIdx.x;
  const int s    = blockIdx.y;
  const int w    = blockIdx.z;
  const int lane = threadIdx.x & 31;
  const int wave = threadIdx.x >> 5;
  const int half = lane >> 4;      // 0: lanes 0-15, 1: lanes 16-31
  const int tl   = lane & 15;

  // V tile kept in native [d][t] order (async copy cannot transpose), padded stride 28.
  __shared__ float ldsV[WAVES * HEAD_DIM * VSTRIDE];   // 57344 B
  __shared__ float ldsP[WAVES * 16 * PSTRIDE];         //  4352 B
  __shared__ float ldsM[WAVES * GQA];
  __shared__ float ldsSE[WAVES * GQA];

  float* Vw = &ldsV[wave * HEAD_DIM * VSTRIDE];
  float* Pw = &ldsP[wave * 16 * PSTRIDE];
  const uint32_t VwOff = (uint32_t)(uintptr_t)Vw;      // wave-relative LDS byte offset

  // P rows >= GQA must stay zero (they feed junk WMMA rows we never read)
  for (int i = lane; i < 16 * PSTRIDE; i += 32) Pw[i] = 0.0f;

  // ---- Q A-fragments (16x4 f32 WMMA A layout), pre-scaled, reused all tiles ----
  const float* qbase = query + ((size_t)s * NUM_HEADS + (size_t)kh * GQA) * HEAD_DIM;
  v2f aq[32];
#pragma unroll
  for (int c = 0; c < 32; ++c) {
    const int d0 = c * 4 + half * 2;
    float x = 0.0f, y = 0.0f;
    if (tl < GQA) {
      x = qbase[tl * HEAD_DIM + d0]     * SCALE;
      y = qbase[tl * HEAD_DIM + d0 + 1] * SCALE;
    }
    aq[c].x = x; aq[c].y = y;
  }

  const int cl = context_lens[w * NUM_SEQS + s];
  const int nt = (cl + BLK - 1) / BLK;                 // active tiles only
  const int* bt = block_tables + ((size_t)w * NUM_SEQS + s) * MAX_BLOCKS;

  float mrow[GQA], serow[GQA];
#pragma unroll
  for (int r = 0; r < GQA; ++r) { mrow[r] = -1e30f; serow[r] = 0.0f; }
  v8f oacc[8];
#pragma unroll
  for (int dt = 0; dt < 8; ++dt) oacc[dt] = (v8f){0,0,0,0,0,0,0,0};

  for (int tile = wave; tile < nt; tile += WAVES) {
    const int nb = bt[tile];
    const size_t blkoff = ((size_t)((size_t)w * NBLOCKS + nb) * NUM_KV_HEADS + kh) * (HEAD_DIM * BLK);
    const float* kb = key_cache   + blkoff;
    const float* vb = value_cache + blkoff;

    // prefetch next tile's K/V (emits global_prefetch_b8); 32 lanes x 256B = 8KB
    if (tile + WAVES < nt) {
      const int nb2 = bt[tile + WAVES];
      const size_t off2 = ((size_t)((size_t)w * NBLOCKS + nb2) * NUM_KV_HEADS + kh) * (HEAD_DIM * BLK);
      __builtin_prefetch(key_cache   + off2 + lane * 64, 0, 1);
      __builtin_prefetch(value_cache + off2 + lane * 64, 0, 1);
    }

    // ---- async-stage V block into LDS (background copy, overlaps QK below) ----
    wait_dscnt0();                                     // prior PV reads must be done before overwrite
#pragma unroll
    for (int i = 0; i < 16; ++i) {
      const int flat = (i * 32 + lane) * 4;            // 4 consecutive t within one d-row
      const int d = flat >> 4, t = flat & 15;
      async_copy_b128(vb + flat, VwOff + (uint32_t)(d * VSTRIDE + t) * 4u);
    }

    // ---- logits tile (16x16): rows 0..3 = queries, cols = 16 timesteps ----
    // K block is stored d-major => B-fragment loads are coalesced from global.
    v8f lg = (v8f){0,0,0,0,0,0,0,0};
#pragma unroll
    for (int c = 0; c < 32; ++c) {
      const int d0 = c * 4 + half * 2;
      v2f b;
      b.x = kb[d0 * BLK + tl];
      b.y = kb[(d0 + 1) * BLK + tl];
      lg = __builtin_amdgcn_wmma_f32_16x16x4_f32(false, aq[c], false, b,
                                                 (short)0, lg, false, false);
    }

    // ---- online softmax per query row ----
    const int  tglob = tile * BLK + tl;
    const bool valid = (half == 0) && (tglob < cl);
    float pbuf[GQA];
#pragma unroll
    for (int r = 0; r < GQA; ++r) {
      const float l      = valid ? lg[r] : -1e30f;
      const float rowmax = wave_max(l);
      const float mnew   = fmaxf(mrow[r], rowmax);
      const float alpha  = __expf(mrow[r] - mnew);
      const float p      = valid ? __expf(l - mnew) : 0.0f;
      serow[r] = serow[r] * alpha + wave_sum(p);
      mrow[r]  = mnew;
      pbuf[r]  = p;
#pragma unroll
      for (int dt = 0; dt < 8; ++dt) oacc[dt][r] *= alpha;   // rescale accumulators
    }

    // ---- transpose P through LDS for the PV A-fragments ----
#pragma unroll
    for (int r = 0; r < GQA; ++r)
      if (half == 0) Pw[r * PSTRIDE + tl] = pbuf[r];

    // V tile must be fully landed before PV reads
    wait_asynccnt0();

    // ---- O(16q x 128d) += P(16x16) * V(16t x 128d), K-dim chunks of 4 ----
#pragma unroll
    for (int kk = 0; kk < 4; ++kk) {
      const int k0 = kk * 4 + half * 2;
      v2f ap;
      ap.x = Pw[tl * PSTRIDE + k0];
      ap.y = Pw[tl * PSTRIDE + k0 + 1];
#pragma unroll
      for (int dt = 0; dt < 8; ++dt) {
        v2f bv;                                        // V[t=k0..][d = dt*16+tl], layout [d][VSTRIDE]
        bv.x = Vw[(dt * 16 + tl) * VSTRIDE + k0];
        bv.y = Vw[(dt * 16 + tl) * VSTRIDE + k0 + 1];
        oacc[dt] = __builtin_amdgcn_wmma_f32_16x16x4_f32(false, ap, false, bv,
                                                         (short)0, oacc[dt], false, false);
      }
    }
  }

  // ---- spill per-wave softmax state + o partials to LDS ----
  // o partials alias this wave's own (now dead) V slice to stay under the LDS budget.
  float* Ow = Vw;
#pragma unroll
  for (int r = 0; r < GQA; ++r) {
    if (lane == 0) { ldsM[wave * GQA + r] = mrow[r]; ldsSE[wave * GQA + r] = serow[r]; }
#pragma unroll
    for (int dt = 0; dt < 8; ++dt)
      if (half == 0) Ow[r * HEAD_DIM + dt * 16 + tl] = oacc[dt][r];
  }
  __syncthreads();

  // ---- merge the 4 wave partials -> per-rank (o_un, m, se) ----
  const int d = threadIdx.x;                           // 0..127
  const size_t sidx = (((size_t)w * NUM_SEQS + s) * NUM_KV_HEADS + kh) * GQA;
#pragma unroll
  for (int r = 0; r < GQA; ++r) {
    float mg = -1e30f;
#pragma unroll
    for (int wv = 0; wv < WAVES; ++wv) mg = fmaxf(mg, ldsM[wv * GQA + r]);
    float den = 0.0f, acc = 0.0f;
#pragma unroll
    for (int wv = 0; wv < WAVES; ++wv) {
      const float al = __expf(ldsM[wv * GQA + r] - mg);
      den += ldsSE[wv * GQA + r] * al;
      acc += ldsV[wv * HEAD_DIM * VSTRIDE + r * HEAD_DIM + d] * al;
    }
    o_un[(sidx + r) * HEAD_DIM + d] = acc;
    if (d == 0) { m_out[sidx + r] = mg; se_out[sidx + r] = den; }
  }
}

// Phase 2: merge per-rank softmax states (distributed_paged_attention_v2_reduce).
__global__ __launch_bounds__(HEAD_DIM)
void pa_reduce_kernel(const float* __restrict__ o_un,
                      const float* __restrict__ m_in,
                      const float* __restrict__ se_in,
                      float* __restrict__ out)
{
  const int g  = blockIdx.x & (GQA - 1);
  const int kh = blockIdx.x >> 2;
  const int s  = blockIdx.y;
  const int d  = threadIdx.x;

  size_t idx[NUM_RANKS];
  float  mg = -1e30f;
#pragma unroll
  for (int w = 0; w < NUM_RANKS; ++w) {
    idx[w] = (((size_t)w * NUM_SEQS + s) * NUM_KV_HEADS + kh) * GQA + g;
    mg = fmaxf(mg, m_in[idx[w]]);
  }
  float den = 0.0f, num = 0.0f;
#pragma unroll
  for (int w = 0; w < NUM_RANKS; ++w) {
    const float al = __expf(m_in[idx[w]] - mg);
    den += se_in[idx[w]] * al;
    num += o_un[idx[w] * HEAD_DIM + d] * al;
  }
  out[((size_t)s * NUM_HEADS + (size_t)kh * GQA + g) * HEAD_DIM + d] = num / den;
}

extern "C" void kernel_launch(void* const* d_in, const int* in_sizes, int n_in,
                              void* d_out, int out_size, void* d_ws, size_t ws_size,
                              hipStream_t stream)
{
  (void)in_sizes; (void)n_in; (void)out_size; (void)ws_size;
  const float* query        = (const float*)d_in[0];
  const float* key_cache    = (const float*)d_in[1];
  const float* value_cache  = (const float*)d_in[2];
  const int*   block_tables = (const int*)d_in[3];
  const int*   context_lens = (const int*)d_in[4];
  float* out = (float*)d_out;

  float* ws    = (float*)d_ws;
  const size_t n_o = (size_t)NUM_RANKS * NUM_SEQS * NUM_KV_HEADS * GQA * HEAD_DIM; // 524288
  const size_t n_m = (size_t)NUM_RANKS * NUM_SEQS * NUM_KV_HEADS * GQA;            // 4096
  float* o_un  = ws;
  float* m_ws  = ws + n_o;
  float* se_ws = m_ws + n_m;

  dim3 g1(NUM_KV_HEADS, NUM_SEQS, NUM_RANKS);
  pa_rank_kernel<<<g1, WAVES * 32, 0, stream>>>(query, key_cache, value_cache,
                                                block_tables, context_lens,
                                                o_un, m_ws, se_ws);

  dim3 g2(NUM_KV_HEADS * GQA, NUM_SEQS, 1);
  pa_reduce_kernel<<<g2, HEAD_DIM, 0, stream>>>(o_un, m_ws, se_ws, out);
}